// Gray_Conv2d_40114994544667
// MI455X (gfx1250) — compile-verified
//
#include <hip/hip_runtime.h>
#include <math.h>

typedef float v2f __attribute__((ext_vector_type(2)));
typedef float v8f __attribute__((ext_vector_type(8)));

constexpr int Bn = 32;
constexpr int Hn = 224;
constexpr int Wn = 224;
constexpr int On = 64;          // output channels
constexpr int Kk = 5;           // conv kernel
constexpr int Dk = 25;          // C*K*K (padded to 28 = 7 chunks of K=4)
constexpr int TILES_W = 14;     // 224 / 16
constexpr int WAVES_PER_BLOCK = 8;
constexpr int LDS_ROW = 20;     // 16 + 4 pad (keeps float4 alignment, spreads banks)

__device__ __forceinline__ float load_x(const float* __restrict__ x, int b, int h, int w) {
    if ((unsigned)h < (unsigned)Hn && (unsigned)w < (unsigned)Wn)
        return x[((size_t)b * Hn + h) * Wn + w];
    return 0.0f;
}

__device__ __forceinline__ float load_w(const float* __restrict__ wt, int n, int k) {
    return (k < Dk) ? wt[n * Dk + k] : 0.0f;
}

__global__ __launch_bounds__(256) void cdist_conv_wmma(
        const float* __restrict__ x,
        const float* __restrict__ wt,
        float* __restrict__ out) {
    __shared__ float lds[WAVES_PER_BLOCK * On * LDS_ROW];

    const int lane      = threadIdx.x & 31;
    const int waveInBlk = threadIdx.x >> 5;
    const int wi        = blockIdx.x * WAVES_PER_BLOCK + waveInBlk;

    const int tilesPerImg = Hn * TILES_W;            // 3136
    const int b  = wi / tilesPerImg;
    const int r0 = wi - b * tilesPerImg;
    const int h  = r0 / TILES_W;
    const int w0 = (r0 - h * TILES_W) * 16;

    const int m    = lane & 15;   // row/col index within 16-tile
    const int half = lane >> 4;   // 0 -> K pair {4c,4c+1}; 1 -> {4c+2,4c+3}

    // ---- B fragments: weight^T (B[k][n] = wt[n][k]), plus |w|^2 partials ----
    v2f   bf[4][7];
    float w2p[4] = {0.f, 0.f, 0.f, 0.f};
    #pragma unroll
    for (int nt = 0; nt < 4; ++nt) {
        const int n = nt * 16 + m;
        #pragma unroll
        for (int c = 0; c < 7; ++c) {
            const int k0 = c * 4 + half * 2;
            v2f bb;
            bb.x = load_w(wt, n, k0);
            bb.y = load_w(wt, n, k0 + 1);
            bf[nt][c] = bb;
            w2p[nt] += bb.x * bb.x + bb.y * bb.y;
        }
    }

    // ---- A fragments: fused im2col (16 positions along W, same image row) ----
    v2f   af[7];
    float x2p = 0.f;
    const int wcol = w0 + m;
    #pragma unroll
    for (int c = 0; c < 7; ++c) {
        const int k0 = c * 4 + half * 2;
        v2f aa;
        {
            const int k  = k0;
            const int kh = k / Kk, kw = k - kh * Kk;
            aa.x = (k < Dk) ? load_x(x, b, h + kh - 2, wcol + kw - 2) : 0.f;
        }
        {
            const int k  = k0 + 1;
            const int kh = k / Kk, kw = k - kh * Kk;
            aa.y = (k < Dk) ? load_x(x, b, h + kh - 2, wcol + kw - 2) : 0.f;
        }
        af[c] = aa;
        x2p += aa.x * aa.x + aa.y * aa.y;
    }
    const float x2 = x2p + __shfl_xor(x2p, 16, 32);  // full 25-elem |patch|^2, in lane (m, m+16)

    // ---- GEMM via fp32 WMMA: 7 K-chunks x 4 N-tiles ----
    v8f acc[4] = {};
    #pragma unroll
    for (int c = 0; c < 7; ++c) {
        #pragma unroll
        for (int nt = 0; nt < 4; ++nt) {
            acc[nt] = __builtin_amdgcn_wmma_f32_16x16x4_f32(
                /*neg_a=*/false, af[c],
                /*neg_b=*/false, bf[nt][c],
                /*c_mod=*/(short)0, acc[nt],
                /*reuse_a=*/false, /*reuse_b=*/false);
        }
    }

    // ---- epilogue: dist = sqrt(max(|x|^2 + |w|^2 - 2 x.w, 0)) -> LDS (o-major) ----
    float* myLds = &lds[waveInBlk * (On * LDS_ROW)];
    #pragma unroll
    for (int nt = 0; nt < 4; ++nt) {
        const float w2 = w2p[nt] + __shfl_xor(w2p[nt], 16, 32); // |w[nt*16+m]|^2
        const int   o  = nt * 16 + m;
        #pragma unroll
        for (int r = 0; r < 8; ++r) {
            const int   mi  = r + half * 8;        // M index held by this acc element
            const float x2m = __shfl(x2, mi, 32);  // |patch mi|^2 lives in lane mi
            const float d2  = x2m + w2 - 2.0f * acc[nt][r];
            myLds[o * LDS_ROW + mi] = sqrtf(fmaxf(d2, 0.0f));
        }
    }

    __syncthreads();

    // ---- coalesced writeout: each o-row (16 floats = one 64B line) via 4x float4 ----
    const int oBase = lane >> 2;       // 0..7
    const int m4    = (lane & 3) * 4;  // 0,4,8,12
    #pragma unroll
    for (int it = 0; it < 8; ++it) {
        const int o = it * 8 + oBase;
        const float4 v = *reinterpret_cast<const float4*>(&myLds[o * LDS_ROW + m4]);
        float* dst = &out[(((size_t)b * On + o) * Hn + h) * Wn + (w0 + m4)];
        *reinterpret_cast<float4*>(dst) = v;
    }
}

extern "C" void kernel_launch(void* const* d_in, const int* in_sizes, int n_in,
                              void* d_out, int out_size, void* d_ws, size_t ws_size,
                              hipStream_t stream) {
    const float* x  = (const float*)d_in[0];   // [32,1,224,224] fp32
    const float* wt = (const float*)d_in[1];   // [64,25] fp32
    float* out = (float*)d_out;                // [32,64,224,224] fp32

    const int totalWaves = Bn * Hn * TILES_W;  // 100352, exact multiple of 8
    dim3 grid(totalWaves / WAVES_PER_BLOCK);   // 12544 blocks
    cdist_conv_wmma<<<grid, 256, 0, stream>>>(x, wt, out);
}